// WRMDGC_56590489092110
// MI455X (gfx1250) — compile-verified
//
#include <hip/hip_runtime.h>

typedef float v2f __attribute__((ext_vector_type(2)));
typedef float v8f __attribute__((ext_vector_type(8)));

__device__ __forceinline__ float sigf(float x){ return 1.0f/(1.0f + __expf(-x)); }

// ---------------- K0: init (zero pads, state <- init_state, gm <- 0) ----------------
__global__ void k_init(const float* init_state, float* state, float* gm, float* nvp, float* xs){
  int id = blockIdx.x*256 + threadIdx.x;
  if (id < 2932736){ state[id] = init_state[id]; gm[id] = 0.f; }
  if (id < 552960)  nvp[id] = 0.f;          // [B,720,12] padded nodevec
  if (id < 3686400) xs[id]  = 0.f;          // [B,720,80] scaled features
}

// ---------------- K1: W[n,row,o] = sum_d Es[n,d]*Wpool[d,row,o]  (row = k*65+i) ----------------
__global__ void k_wpool(const float* Es, const float* Wpool, float* Wg, int dout, int total){
  int id = blockIdx.x*256 + threadIdx.x;
  if (id >= total) return;
  int per = 130*dout;
  int n = id / per, rd = id % per;
  float s = 0.f;
#pragma unroll
  for (int d=0; d<10; ++d) s += Es[n*10+d]*Wpool[d*per+rd];
  Wg[id] = s;
}
__global__ void k_bpool(const float* Es, const float* bpool, float* bg, int dout, int total){
  int id = blockIdx.x*256 + threadIdx.x;
  if (id >= total) return;
  int n = id/dout, o = id%dout;
  float s = 0.f;
#pragma unroll
  for (int d=0; d<10; ++d) s += Es[n*10+d]*bpool[d*dout+o];
  bg[id] = s;
}

// ---------------- K2: per (b,t): mean_n |x-ref| for weekly/daily/recent ----------------
__global__ void k_ssum(const float* x, const float* wk, const float* dy, const float* rc, float* sbuf){
  __shared__ float red[3*256];
  int bt = blockIdx.x, tid = threadIdx.x;
  float s0=0.f, s1=0.f, s2=0.f;
  for (int n=tid; n<716; n+=256){
    float xv = x[bt*716+n];
    s0 += fabsf(xv - wk[bt*716+n]);
    s1 += fabsf(xv - dy[bt*716+n]);
    s2 += fabsf(xv - rc[bt*716+n]);
  }
  red[tid]=s0; red[256+tid]=s1; red[512+tid]=s2;
  __syncthreads();
  for (int st=128; st>0; st>>=1){
    if (tid<st){
      red[tid]      += red[tid+st];
      red[256+tid]  += red[256+tid+st];
      red[512+tid]  += red[512+tid+st];
    }
    __syncthreads();
  }
  if (tid==0){
    sbuf[bt*3+0]=red[0]/716.f;
    sbuf[bt*3+1]=red[256]/716.f;
    sbuf[bt*3+2]=red[512]/716.f;
  }
}

// ---------------- K3: cur = x*(1 + sum_ref w_ref[b,t]) ----------------
__global__ void k_cur(const float* x, const float* sbuf, float* cur){
  int id = blockIdx.x*256 + threadIdx.x;
  if (id >= 549888) return;
  int b = id/(12*716);
  int rem = id%(12*716);
  int t = rem/716;
  float wsum = 0.f;
#pragma unroll
  for (int k=0;k<3;++k){
    float den = 1e-6f;
#pragma unroll
    for (int tt=0;tt<12;++tt) den += sbuf[(b*12+tt)*3+k];
    wsum += sbuf[(b*12+t)*3+k]/den;
  }
  cur[id] = x[id]*(1.f+wsum);
}

// ---------------- K4: time-weight MLP: sigmoid(relu(relu([t,d]W1+b)W2+b)W3+b) ----------------
__global__ void k_tw(const float* timei, const float* dayi,
                     const float* Wt1, const float* bt1,
                     const float* Wt2, const float* bt2,
                     const float* Wt3, const float* bt3, float* tw){
  int id = blockIdx.x*256 + threadIdx.x;
  if (id >= 549888) return;
  float in0 = timei[id], in1 = dayi[id];
  float h2a = bt2[0], h2b = bt2[1];
#pragma unroll 1
  for (int j=0;j<40;++j){
    float h = fmaxf(in0*Wt1[j] + in1*Wt1[40+j] + bt1[j], 0.f);
    h2a += h*Wt2[j*2+0];
    h2b += h*Wt2[j*2+1];
  }
  h2a = fmaxf(h2a,0.f); h2b = fmaxf(h2b,0.f);
  tw[id] = sigf(h2a*Wt3[0] + h2b*Wt3[1] + bt3[0]);
}

// ---------------- K5: hyper-MLP -> nodevec; also builds xin = [cur_t, (z*)state] ----------------
__global__ void k_mlp(const float* cur, const float* state, const float* zr, int mode, int t,
                      const float* W1, const float* b1, const float* W2, const float* b2,
                      const float* W3, const float* b3, const float* Ed,
                      float* xin, float* nvp){
  __shared__ float lW1[4225], lW2[4225], lW3[650], lb[140];
  int tid = threadIdx.x;
  for (int k=tid;k<4225;k+=256){ lW1[k]=W1[k]; lW2[k]=W2[k]; }
  for (int k=tid;k<650;k+=256) lW3[k]=W3[k];
  if (tid<65) lb[tid]=b1[tid];
  if (tid>=65 && tid<130) lb[tid]=b2[tid-65];
  if (tid>=130 && tid<140) lb[tid]=b3[tid-130];
  __syncthreads();
  int id = blockIdx.x*256 + tid;
  if (id >= 64*716) return;
  int b = id/716, n = id%716;
  const int sb = b*716 + n;
  float acc1[65];
#pragma unroll
  for (int j=0;j<65;++j) acc1[j] = lb[j];
#pragma unroll 1
  for (int i=0;i<65;++i){
    float xi;
    if (i==0) xi = cur[(b*12+t)*716+n];
    else {
      float sv = state[(size_t)sb*64 + i - 1];
      if (mode==1) sv *= zr[(size_t)sb*128 + i - 1];
      xi = sv;
    }
    xin[(size_t)sb*65 + i] = xi;
#pragma unroll
    for (int j=0;j<65;++j) acc1[j] += xi * lW1[i*65+j];
  }
  float h1[65];
#pragma unroll
  for (int j=0;j<65;++j) h1[j] = sigf(acc1[j]);
  float e[10];
#pragma unroll
  for (int d=0;d<10;++d) e[d] = lb[130+d];
#pragma unroll 1
  for (int j=0;j<65;++j){
    float ta = lb[65+j];
#pragma unroll
    for (int i=0;i<65;++i) ta += h1[i]*lW2[i*65+j];
    ta = sigf(ta);
#pragma unroll
    for (int d=0;d<10;++d) e[d] += ta * lW3[j*10+d];
  }
#pragma unroll
  for (int d=0;d<10;++d)
    nvp[((size_t)b*720+n)*12 + d] = tanhf(Ed[(((size_t)b*12+t)*716+n)*10 + d] * e[d]);
}

// ---------------- K6: dinv = rsqrt(1 + sum_m relu(nv_n . nv_m)); xs = dinv*xin ----------------
__global__ void k_dinv(const float* nvp, const float* xin, float* dinv, float* xs){
  __shared__ float nvb[7160];
  int b = blockIdx.y, tid = threadIdx.x;
  for (int k=tid;k<7160;k+=256){ int m=k/10, d=k%10; nvb[k]=nvp[((size_t)b*720+m)*12+d]; }
  __syncthreads();
  int n = blockIdx.x*256 + tid;
  if (n >= 716) return;
  float my[10];
#pragma unroll
  for (int d=0;d<10;++d) my[d]=nvb[n*10+d];
  float sum = 1.0f;                       // +I diagonal
#pragma unroll 1
  for (int m=0;m<716;++m){
    float dot = 0.f;
#pragma unroll
    for (int d=0;d<10;++d) dot += my[d]*nvb[m*10+d];
    sum += fmaxf(dot, 0.f);
  }
  float dv = rsqrtf(sum);
  dinv[b*720+n] = dv;
  const float* xr = xin + ((size_t)b*716+n)*65;
  float* xo = xs + ((size_t)b*720+n)*80;
#pragma unroll 1
  for (int c=0;c<65;++c) xo[c] = dv * xr[c];
}

// ---------------- K7: y = dinv_r * ( relu(nv nv^T) @ (dinv*x) + dinv_r*x_r )  [WMMA f32] ----------------
__global__ void k_lapx(const float* nvp, const float* xs, const float* dinv, float* y){
  __shared__ float ldsA[256];
  int tid = threadIdx.x;
  int wave = tid>>5, lane = tid&31, lh = lane>>4, ll = lane&15;
  int b = blockIdx.y, rbase = blockIdx.x*16;
  const float* nvb = nvp + (size_t)b*720*12;
  const float* xsb = xs  + (size_t)b*720*80;
  int ct = wave;                           // 5 waves = 5 column tiles (80 cols)
  v8f acc = {0.f,0.f,0.f,0.f,0.f,0.f,0.f,0.f};
#pragma unroll 1
  for (int mb=0; mb<720; mb+=16){
    // A tile: nv[rbase..+15] x nv[mb..+15]^T, K=12 -> 3 WMMAs
    v8f at = {0.f,0.f,0.f,0.f,0.f,0.f,0.f,0.f};
#pragma unroll
    for (int s=0;s<3;++s){
      int k = 4*s + 2*lh;
      v2f av, bv;
      av.x = nvb[(rbase+ll)*12 + k];
      av.y = nvb[(rbase+ll)*12 + k + 1];
      bv.x = nvb[(mb+ll)*12 + k];
      bv.y = nvb[(mb+ll)*12 + k + 1];
      at = __builtin_amdgcn_wmma_f32_16x16x4_f32(false, av, false, bv, (short)0, at, false, false);
    }
#pragma unroll
    for (int v=0;v<8;++v) at[v] = fmaxf(at[v], 0.f);   // relu
    __syncthreads();
    if (wave==0){
#pragma unroll
      for (int v=0;v<8;++v) ldsA[(v + 8*lh)*16 + ll] = at[v];  // D-layout -> LDS
    }
    __syncthreads();
    // acc += Atile(16x16) x xs[mb..+15, ct*16..+15], K=16 -> 4 WMMAs
#pragma unroll
    for (int s=0;s<4;++s){
      int k = 4*s + 2*lh;
      v2f av, bv;
      av.x = ldsA[ll*16 + k];
      av.y = ldsA[ll*16 + k + 1];
      bv.x = xsb[(mb + k)*80 + ct*16 + ll];
      bv.y = xsb[(mb + k + 1)*80 + ct*16 + ll];
      acc = __builtin_amdgcn_wmma_f32_16x16x4_f32(false, av, false, bv, (short)0, acc, false, false);
    }
  }
#pragma unroll
  for (int v=0;v<8;++v){
    int r = rbase + v + 8*lh;
    int c = ct*16 + ll;
    if (r < 716 && c < 65){
      float val = dinv[b*720+r] * (acc[v] + xsb[r*80 + c]);   // +I term
      y[((size_t)b*716 + r)*65 + c] = val;
    }
  }
}

// ---------------- K8: per node n: out[b,o] = act( xg[64x130] @ W_n[130xdout] + bias )  [WMMA f32] ----------------
__global__ void k_dgout(const float* xin, const float* ybuf, const float* W,
                        const float* bias, float* outp, int dout, int act){
  extern __shared__ float sm[];
  float* xgL = sm;            // 64 x 132 (K padded with zeros)
  float* WL  = sm + 64*132;   // 132 x dout
  int n = blockIdx.x, tid = threadIdx.x;
  for (int idx=tid; idx<132*dout; idx+=256){
    int row = idx/dout, o = idx%dout;
    WL[idx] = (row<130) ? W[((size_t)n*130 + row)*dout + o] : 0.f;
  }
  for (int idx=tid; idx<64*132; idx+=256){
    int b = idx/132, i = idx%132;
    float v;
    if (i<65)       v = xin [((size_t)b*716+n)*65 + i];
    else if (i<130) v = ybuf[((size_t)b*716+n)*65 + (i-65)];
    else            v = 0.f;
    xgL[idx] = v;
  }
  __syncthreads();
  int wave = tid>>5, lane = tid&31, lh = lane>>4, ll = lane&15;
  int nct = dout>>4;
  int ntiles = 4*nct;          // 4 row tiles (64 batches) x col tiles
#pragma unroll 1
  for (int tile=wave; tile<ntiles; tile+=8){
    int rt = tile/nct, ctt = tile%nct;
    v8f acc = {0.f,0.f,0.f,0.f,0.f,0.f,0.f,0.f};
#pragma unroll 1
    for (int s=0;s<33;++s){
      int k = 4*s + 2*lh;
      v2f av, bv;
      av.x = xgL[(rt*16+ll)*132 + k];
      av.y = xgL[(rt*16+ll)*132 + k + 1];
      bv.x = WL[k*dout + ctt*16 + ll];
      bv.y = WL[(k+1)*dout + ctt*16 + ll];
      acc = __builtin_amdgcn_wmma_f32_16x16x4_f32(false, av, false, bv, (short)0, acc, false, false);
    }
#pragma unroll
    for (int v=0;v<8;++v){
      int bb = rt*16 + v + 8*lh;
      int o  = ctt*16 + ll;
      float val = acc[v] + bias[n*dout + o];
      val = (act==0) ? sigf(val) : tanhf(val);
      outp[((size_t)bb*716 + n)*dout + o] = val;
    }
  }
}

// ---------------- K9: GRU combine + memory update + write output ----------------
__global__ void k_combine(const float* zr, const float* hc, const float* tw,
                          const float* a_p, const float* b_p, int t,
                          float* state, float* gm, float* outp){
  int id = blockIdx.x*256 + threadIdx.x;
  if (id >= 2932736) return;
  int b = id/(716*64);
  int rem = id%(716*64);
  int n = rem/64, h = rem%64;
  float r   = zr[((size_t)b*716+n)*128 + 64 + h];
  float hcv = hc[id];
  float so  = state[id];
  float sn  = r*so + (1.f-r)*hcv;
  float twv = tw[(b*12+t)*716+n];
  float gmo = gm[id];
  float enh = twv*sn + (1.f-twv)*gmo;
  float av = a_p[0], bv = b_p[0];
  gm[id] = (1.f-av)*gmo + av*enh;
  float sf = sn + bv*enh;
  state[id] = sf;
  outp[(((size_t)b*12+t)*716+n)*64 + h] = sf;
}

// ---------------- K10: out_hidden ----------------
__global__ void k_copy(const float* state, float* outp){
  int id = blockIdx.x*256 + threadIdx.x;
  if (id < 2932736) outp[id] = state[id];
}

extern "C" void kernel_launch(void* const* d_in, const int* in_sizes, int n_in,
                              void* d_out, int out_size, void* d_ws, size_t ws_size,
                              hipStream_t stream){
  (void)in_sizes; (void)n_in; (void)out_size; (void)ws_size;
  const float* x          = (const float*)d_in[0];
  const float* init_state = (const float*)d_in[1];
  const float* Ed         = (const float*)d_in[2];
  const float* Es         = (const float*)d_in[3];
  const float* timei      = (const float*)d_in[4];
  const float* dayi       = (const float*)d_in[5];
  const float* weekly     = (const float*)d_in[8];
  const float* daily      = (const float*)d_in[9];
  const float* recent     = (const float*)d_in[10];
  // params pytree (dicts flattened in sorted-key order): a, b, gate(...), twg(...), update(...)
  const float* a_p = (const float*)d_in[11];
  const float* b_p = (const float*)d_in[12];
  const float* gW1=(const float*)d_in[13], *gW2=(const float*)d_in[14], *gW3=(const float*)d_in[15];
  const float* gWpool=(const float*)d_in[16];
  const float* gb1=(const float*)d_in[17], *gb2=(const float*)d_in[18], *gb3=(const float*)d_in[19];
  const float* gbpool=(const float*)d_in[20];
  const float* Wt1=(const float*)d_in[21], *Wt2=(const float*)d_in[22], *Wt3=(const float*)d_in[23];
  const float* bt1=(const float*)d_in[24], *bt2=(const float*)d_in[25], *bt3=(const float*)d_in[26];
  const float* uW1=(const float*)d_in[27], *uW2=(const float*)d_in[28], *uW3=(const float*)d_in[29];
  const float* uWpool=(const float*)d_in[30];
  const float* ub1=(const float*)d_in[31], *ub2=(const float*)d_in[32], *ub3=(const float*)d_in[33];
  const float* ubpool=(const float*)d_in[34];

  float* ws = (float*)d_ws;
  size_t off = 0;
  float* cur   = ws + off; off += 549888;    // [B,T,N]
  float* state = ws + off; off += 2932736;   // [B,N,H]
  float* gm    = ws + off; off += 2932736;   // [B,N,H]
  float* tw    = ws + off; off += 549888;    // [B,T,N]
  float* sbuf  = ws + off; off += 2304;      // [B,T,3]
  float* nvp   = ws + off; off += 552960;    // [B,720,12]
  float* dinvb = ws + off; off += 46080;     // [B,720]
  float* xin   = ws + off; off += 2978560;   // [B,716,65]
  float* xs    = ws + off; off += 3686400;   // [B,720,80]
  float* ybuf  = ws + off; off += 2978560;   // [B,716,65]
  float* zr    = ws + off; off += 5865472;   // [B,716,128]
  float* hc    = ws + off; off += 2932736;   // [B,716,64]
  float* Wg    = ws + off; off += 11914240;  // [716,130,128]
  float* Wu    = ws + off; off += 5957120;   // [716,130,64]
  float* bg    = ws + off; off += 91648;     // [716,128]
  float* bu    = ws + off; off += 45824;     // [716,64]

  float* outp = (float*)d_out;

  k_init <<<14400,256,0,stream>>>(init_state, state, gm, nvp, xs);
  k_wpool<<<(11914240+255)/256,256,0,stream>>>(Es, gWpool, Wg, 128, 11914240);
  k_wpool<<<(5957120+255)/256,256,0,stream>>>(Es, uWpool, Wu, 64, 5957120);
  k_bpool<<<(91648+255)/256,256,0,stream>>>(Es, gbpool, bg, 128, 91648);
  k_bpool<<<(45824+255)/256,256,0,stream>>>(Es, ubpool, bu, 64, 45824);
  k_ssum <<<768,256,0,stream>>>(x, weekly, daily, recent, sbuf);
  k_cur  <<<2148,256,0,stream>>>(x, sbuf, cur);
  k_tw   <<<2148,256,0,stream>>>(timei, dayi, Wt1, bt1, Wt2, bt2, Wt3, bt3, tw);

  for (int t=0; t<12; ++t){
    // gate dgcn -> zr = sigmoid(...)
    k_mlp  <<<179,256,0,stream>>>(cur, state, zr, 0, t, gW1,gb1,gW2,gb2,gW3,gb3, Ed, xin, nvp);
    k_dinv <<<dim3(3,64),256,0,stream>>>(nvp, xin, dinvb, xs);
    k_lapx <<<dim3(45,64),160,0,stream>>>(nvp, xs, dinvb, ybuf);
    k_dgout<<<716,256,(64*132+132*128)*sizeof(float),stream>>>(xin, ybuf, Wg, bg, zr, 128, 0);
    // update dgcn -> hc = tanh(...)
    k_mlp  <<<179,256,0,stream>>>(cur, state, zr, 1, t, uW1,ub1,uW2,ub2,uW3,ub3, Ed, xin, nvp);
    k_dinv <<<dim3(3,64),256,0,stream>>>(nvp, xin, dinvb, xs);
    k_lapx <<<dim3(45,64),160,0,stream>>>(nvp, xs, dinvb, ybuf);
    k_dgout<<<716,256,(64*132+132*64)*sizeof(float),stream>>>(xin, ybuf, Wu, bu, hc, 64, 1);
    // GRU combine + time-enhanced memory
    k_combine<<<11456,256,0,stream>>>(zr, hc, tw, a_p, b_p, t, state, gm, outp);
  }
  k_copy<<<11456,256,0,stream>>>(state, outp + 35192832);
}